// SingleGAT_9457517986121
// MI455X (gfx1250) — compile-verified
//
#include <hip/hip_runtime.h>
#include <hip/hip_bf16.h>
#include <math.h>

#define N_NODES   50000
#define N_EDGES   800000
#define TOT_E     (N_EDGES + N_NODES)   // edges + self loops
#define HEADS     4
#define HC        256                    // HEADS * OUT_CH
#define IN_CH     64
#define NEG_SLOPE 0.2f

typedef __attribute__((ext_vector_type(2))) float v2f;
typedef __attribute__((ext_vector_type(8))) float v8f;

// ---------------------------------------------------------------------------
// Kernel 1: xt = x @ W  (M=50000, N=256, K=64) in fp32 via V_WMMA_F32_16X16X4_F32
// Block = 256 threads = 8 waves; block computes 16 rows x 256 cols.
// Wave w computes column tiles 2w and 2w+1 (16 cols each).
// A tile (16x64 fp32) staged in LDS; B fragments streamed from L2-resident W.
// ---------------------------------------------------------------------------
__global__ __launch_bounds__(256)
void gat_gemm_wmma(const float* __restrict__ x, const float* __restrict__ W,
                   float* __restrict__ xt) {
  __shared__ float As[16 * IN_CH];           // 4 KB
  const int row0 = blockIdx.x * 16;
  const int t = threadIdx.x;

  // Cooperative A-tile load: 1024 floats / 256 threads = 1 float4 each.
  {
    const int r = t >> 4;                    // 0..15
    const int c = (t & 15) * 4;              // 0..60
    const float4 v = *(const float4*)(x + (size_t)(row0 + r) * IN_CH + c);
    *(float4*)(As + r * IN_CH + c) = v;
  }
  __syncthreads();

  const int wave = t >> 5;                   // 0..7
  const int lane = t & 31;
  const int lhi  = lane >> 4;                // 0|1 : which K-half / M-half
  const int llo  = lane & 15;

  #pragma unroll
  for (int sub = 0; sub < 2; ++sub) {
    const int ncol0 = (wave * 2 + sub) * 16; // tile column base (0..240)
    v8f acc = {};
    #pragma unroll
    for (int k = 0; k < 16; ++k) {           // K = 64, 4 per WMMA
      const int kb = k * 4 + lhi * 2;
      // A fragment (16x4 f32): lane holds M=llo, K=kb..kb+1  -> ds_load_b64
      const v2f a = *(const v2f*)(As + llo * IN_CH + kb);
      // B fragment (4x16 f32): lane holds K=kb..kb+1, N=ncol0+llo
      v2f b;
      b.x = W[(size_t)kb       * HC + ncol0 + llo];
      b.y = W[(size_t)(kb + 1) * HC + ncol0 + llo];
      acc = __builtin_amdgcn_wmma_f32_16x16x4_f32(false, a, false, b,
                                                  (short)0, acc, false, false);
    }
    // C/D layout: VGPR v -> M = v + 8*lhi, N = llo
    #pragma unroll
    for (int v = 0; v < 8; ++v) {
      xt[(size_t)(row0 + v + 8 * lhi) * HC + ncol0 + llo] = acc[v];
    }
  }
}

// ---------------------------------------------------------------------------
// Kernel 2: a_src[n,h] = xt[n,h,:].att_src[h], a_dst likewise. Wave per node.
// ---------------------------------------------------------------------------
__global__ __launch_bounds__(256)
void gat_att_dots(const float* __restrict__ xt,
                  const float* __restrict__ att_src,
                  const float* __restrict__ att_dst,
                  float* __restrict__ asrc, float* __restrict__ adst) {
  const int wave = threadIdx.x >> 5;
  const int lane = threadIdx.x & 31;
  const int n = blockIdx.x * 8 + wave;
  if (n >= N_NODES) return;
  const float* row = xt + (size_t)n * HC;
  #pragma unroll
  for (int h = 0; h < HEADS; ++h) {
    const float x0 = row[h * 64 + lane];
    const float x1 = row[h * 64 + lane + 32];
    float s = x0 * att_src[h * 64 + lane] + x1 * att_src[h * 64 + lane + 32];
    float d = x0 * att_dst[h * 64 + lane] + x1 * att_dst[h * 64 + lane + 32];
    #pragma unroll
    for (int off = 16; off > 0; off >>= 1) {
      s += __shfl_xor(s, off, 32);
      d += __shfl_xor(d, off, 32);
    }
    if (lane == 0) { asrc[n * HEADS + h] = s; adst[n * HEADS + h] = d; }
  }
}

// ---------------------------------------------------------------------------
// Kernel 3: init m=-inf, s=0, acc=0 (ws is poisoned by harness)
// ---------------------------------------------------------------------------
__global__ void gat_init(float* __restrict__ m, float* __restrict__ s,
                         float* __restrict__ acc) {
  const size_t i = (size_t)blockIdx.x * blockDim.x + threadIdx.x;
  if (i < (size_t)N_NODES * HEADS) { m[i] = -INFINITY; s[i] = 0.0f; }
  if (i < (size_t)N_NODES * HC)    acc[i] = 0.0f;
}

// float atomic max via signed/unsigned integer ordering trick
__device__ __forceinline__ void atomicMaxF(float* addr, float val) {
  if (val >= 0.0f) atomicMax((int*)addr, __float_as_int(val));
  else             atomicMin((unsigned int*)addr, __float_as_uint(val));
}

// ---------------------------------------------------------------------------
// Kernel 4: e = leaky_relu(a_src[src]+a_dst[dst]); segment max into m[dst]
// ---------------------------------------------------------------------------
__global__ __launch_bounds__(256)
void gat_edge_logits(const int* __restrict__ ei,
                     const float* __restrict__ asrc,
                     const float* __restrict__ adst,
                     float* __restrict__ ebuf, float* __restrict__ m) {
  const int e = blockIdx.x * blockDim.x + threadIdx.x;
  if (e >= TOT_E) return;
  int src, dst;
  if (e < N_EDGES) { src = ei[e]; dst = ei[N_EDGES + e]; }
  else             { src = dst = e - N_EDGES; }               // self loop
  #pragma unroll
  for (int h = 0; h < HEADS; ++h) {
    float v = asrc[src * HEADS + h] + adst[dst * HEADS + h];
    v = (v > 0.0f) ? v : NEG_SLOPE * v;
    ebuf[(size_t)e * HEADS + h] = v;
    atomicMaxF(&m[dst * HEADS + h], v);
  }
}

// ---------------------------------------------------------------------------
// Kernel 5: e = exp(e - m[dst]); segment sum into s[dst]
// ---------------------------------------------------------------------------
__global__ __launch_bounds__(256)
void gat_edge_exp(const int* __restrict__ ei, const float* __restrict__ m,
                  float* __restrict__ ebuf, float* __restrict__ s) {
  const int e = blockIdx.x * blockDim.x + threadIdx.x;
  if (e >= TOT_E) return;
  const int dst = (e < N_EDGES) ? ei[N_EDGES + e] : (e - N_EDGES);
  #pragma unroll
  for (int h = 0; h < HEADS; ++h) {
    const float v = __expf(ebuf[(size_t)e * HEADS + h] - m[dst * HEADS + h]);
    ebuf[(size_t)e * HEADS + h] = v;
    atomicAdd(&s[dst * HEADS + h], v);
  }
}

// ---------------------------------------------------------------------------
// Kernel 6: acc[dst,h,c] += (e/s[dst,h]) * xt[src,h,c].  Wave per edge;
// each lane handles two float4 chunks (c = 4*lane and 128 + 4*lane).
// ---------------------------------------------------------------------------
__global__ __launch_bounds__(256)
void gat_scatter(const int* __restrict__ ei, const float* __restrict__ xt,
                 const float* __restrict__ ebuf, const float* __restrict__ s,
                 float* __restrict__ acc) {
  const int wave = threadIdx.x >> 5;
  const int lane = threadIdx.x & 31;
  const int e = blockIdx.x * 8 + wave;
  if (e >= TOT_E) return;
  int src, dst;
  if (e < N_EDGES) { src = ei[e]; dst = ei[N_EDGES + e]; }
  else             { src = dst = e - N_EDGES; }
  #pragma unroll
  for (int half = 0; half < 2; ++half) {
    const int cc = lane * 4 + half * 128;       // 0..252
    const int h  = cc >> 6;
    const float alpha = ebuf[(size_t)e * HEADS + h] / s[dst * HEADS + h];
    const float4 v = *(const float4*)(xt + (size_t)src * HC + cc);
    float* p = acc + (size_t)dst * HC + cc;
    atomicAdd(p + 0, alpha * v.x);
    atomicAdd(p + 1, alpha * v.y);
    atomicAdd(p + 2, alpha * v.z);
    atomicAdd(p + 3, alpha * v.w);
  }
}

// ---------------------------------------------------------------------------
// Kernel 7: head mean + bias + relu + fc.  Wave per node.
// ---------------------------------------------------------------------------
__global__ __launch_bounds__(256)
void gat_finalize(const float* __restrict__ acc, const float* __restrict__ bias,
                  const float* __restrict__ fc_w, const float* __restrict__ fc_b,
                  float* __restrict__ out) {
  const int wave = threadIdx.x >> 5;
  const int lane = threadIdx.x & 31;
  const int n = blockIdx.x * 8 + wave;
  if (n >= N_NODES) return;
  const float* row = acc + (size_t)n * HC;
  float r = 0.0f;
  #pragma unroll
  for (int j = 0; j < 2; ++j) {
    const int ch = lane + 32 * j;               // 0..63
    float v = 0.0f;
    #pragma unroll
    for (int h = 0; h < HEADS; ++h) v += row[h * 64 + ch];
    v = v * 0.25f + bias[ch];                   // mean over 4 heads + bias
    v = fmaxf(v, 0.0f);                         // relu
    r += v * fc_w[ch];
  }
  #pragma unroll
  for (int off = 16; off > 0; off >>= 1) r += __shfl_xor(r, off, 32);
  if (lane == 0) out[n] = r + fc_b[0];
}

// ---------------------------------------------------------------------------
extern "C" void kernel_launch(void* const* d_in, const int* in_sizes, int n_in,
                              void* d_out, int out_size, void* d_ws, size_t ws_size,
                              hipStream_t stream) {
  const float* x       = (const float*)d_in[0];
  const int*   ei      = (const int*)d_in[1];   // [2, E] flat: src row, then dst row
  const float* W       = (const float*)d_in[2];
  const float* att_src = (const float*)d_in[3];
  const float* att_dst = (const float*)d_in[4];
  const float* bias    = (const float*)d_in[5];
  const float* fc_w    = (const float*)d_in[6];
  const float* fc_b    = (const float*)d_in[7];
  float* out = (float*)d_out;

  // workspace layout (fp32): ~119 MB total
  float* ws   = (float*)d_ws;
  float* xt   = ws;  ws += (size_t)N_NODES * HC;      // 12.8M
  float* asrc = ws;  ws += (size_t)N_NODES * HEADS;   // 200K
  float* adst = ws;  ws += (size_t)N_NODES * HEADS;
  float* m    = ws;  ws += (size_t)N_NODES * HEADS;
  float* s    = ws;  ws += (size_t)N_NODES * HEADS;
  float* ebuf = ws;  ws += (size_t)TOT_E * HEADS;     // 3.4M
  float* acc  = ws;  ws += (size_t)N_NODES * HC;      // 12.8M

  gat_gemm_wmma <<<N_NODES / 16, 256, 0, stream>>>(x, W, xt);
  gat_att_dots  <<<N_NODES / 8, 256, 0, stream>>>(xt, att_src, att_dst, asrc, adst);
  gat_init      <<<((size_t)N_NODES * HC + 255) / 256, 256, 0, stream>>>(m, s, acc);
  gat_edge_logits<<<(TOT_E + 255) / 256, 256, 0, stream>>>(ei, asrc, adst, ebuf, m);
  gat_edge_exp  <<<(TOT_E + 255) / 256, 256, 0, stream>>>(ei, m, ebuf, s);
  gat_scatter   <<<(TOT_E + 7) / 8, 256, 0, stream>>>(ei, xt, ebuf, s, acc);
  gat_finalize  <<<N_NODES / 8, 256, 0, stream>>>(acc, bias, fc_w, fc_b, out);
}